// SphericalHarmonics_78142634983510
// MI455X (gfx1250) — compile-verified
//
#include <hip/hip_runtime.h>
#include <hip/hip_bf16.h>
#include <stdint.h>

// Spherical harmonics (l<=3) evaluation, N = 2,000,000 points.
// Memory-bound (~0.7 FLOP/byte): optimized around the gfx1250 async
// global->LDS copy path (ASYNCcnt) so the 180 B/point coefficient records
// stream as dense, aligned b128 transfers, then are consumed from LDS at
// odd strides (bank-conflict-free on 64 banks).

#define PTS_PER_BLOCK 256
#define BLOCK_THREADS 256

__global__ __launch_bounds__(BLOCK_THREADS)
void SphericalHarmonics_kernel(const float* __restrict__ dirs,   // (N,2)
                               const float* __restrict__ dc,     // (N,3,1)
                               const float* __restrict__ rest,   // (N,3,15)
                               float* __restrict__ out,          // (N,3)
                               int N)
{
    // LDS layout: [0,768) floats = dc (256*3), [768,12288) = rest (256*45)
    __shared__ float smem[PTS_PER_BLOCK * 48];   // 48 KB

    const int tid        = threadIdx.x;
    const int blockStart = blockIdx.x * PTS_PER_BLOCK;
    int pts = N - blockStart;
    if (pts > PTS_PER_BLOCK) pts = PTS_PER_BLOCK;

    // Raw LDS byte addresses (addrspacecast(3->0) low 32 bits == LDS offset)
    const unsigned ldsBase  = (unsigned)(uintptr_t)(&smem[0]);
    const unsigned dcLds    = ldsBase;
    const unsigned restLds  = ldsBase + 768u * 4u;

    const char* dcG   = (const char*)(dc   + (size_t)blockStart * 3);
    const char* restG = (const char*)(rest + (size_t)blockStart * 45);

    // 16-byte chunk counts (pts is 256 or 128 -> both exact)
    const int dcChunks   = (pts * 3)  >> 2;   // pts*12/16
    const int restChunks = (pts * 45) >> 2;   // pts*180/16

    // Async global->LDS staging: each lane moves one dense b128 chunk.
    for (int c = tid; c < dcChunks; c += BLOCK_THREADS) {
        unsigned    l = dcLds + ((unsigned)c << 4);
        const char* g = dcG   + ((size_t)c << 4);
        asm volatile("global_load_async_to_lds_b128 %0, %1, off"
                     :: "v"(l), "v"(g) : "memory");
    }
    for (int c = tid; c < restChunks; c += BLOCK_THREADS) {
        unsigned    l = restLds + ((unsigned)c << 4);
        const char* g = restG   + ((size_t)c << 4);
        asm volatile("global_load_async_to_lds_b128 %0, %1, off"
                     :: "v"(l), "v"(g) : "memory");
    }
    asm volatile("s_wait_asynccnt 0" ::: "memory");
    __syncthreads();

    if (tid < pts) {
        const int p = blockStart + tid;

        // directions: coalesced b64 straight from global
        const float2 d = ((const float2*)dirs)[p];   // (theta, phi)
        const float st = __sinf(d.x), ct = __cosf(d.x);
        const float sp = __sinf(d.y), cp = __cosf(d.y);
        const float x = st * cp, y = st * sp, z = ct;
        const float xx = x*x, yy = y*y, zz = z*z;
        const float xy = x*y, yz = y*z, xz = x*z;

        float Y[16];
        Y[0]  = 0.28209479177387814f;
        Y[1]  = -0.4886025119029199f * y;
        Y[2]  =  0.4886025119029199f * z;
        Y[3]  = -0.4886025119029199f * x;
        Y[4]  =  1.0925484305920792f * xy;
        Y[5]  = -1.0925484305920792f * yz;
        Y[6]  =  0.31539156525252005f * (2.0f*zz - xx - yy);
        Y[7]  = -1.0925484305920792f * xz;
        Y[8]  =  0.5462742152960396f * (xx - yy);
        Y[9]  = -0.5900435899266435f * y * (3.0f*xx - yy);
        Y[10] =  2.890611442640554f  * xy * z;
        Y[11] = -0.4570457994644658f * y * (4.0f*zz - xx - yy);
        Y[12] =  0.3731763325901154f * z * (2.0f*zz - 3.0f*xx - 3.0f*yy);
        Y[13] = -0.4570457994644658f * x * (4.0f*zz - xx - yy);
        Y[14] =  1.445305721320277f  * z * (xx - yy);
        Y[15] = -0.5900435899266435f * x * (xx - 3.0f*yy);

        // LDS reads: stride 3 / stride 45 across lanes -> conflict-free
        const float* sdc = &smem[0]   + tid * 3;
        const float* sr  = &smem[768] + tid * 45;

        #pragma unroll
        for (int c = 0; c < 3; ++c) {
            float acc = sdc[c] * Y[0];
            #pragma unroll
            for (int i = 1; i < 16; ++i)
                acc = fmaf(sr[c * 15 + (i - 1)], Y[i], acc);
            out[(size_t)p * 3 + c] = acc;
        }
    }
}

extern "C" void kernel_launch(void* const* d_in, const int* in_sizes, int n_in,
                              void* d_out, int out_size, void* d_ws, size_t ws_size,
                              hipStream_t stream) {
    const float* dirs = (const float*)d_in[0];   // (N,2)
    const float* dc   = (const float*)d_in[1];   // (N,3,1)
    const float* rest = (const float*)d_in[2];   // (N,3,15)
    float*       out  = (float*)d_out;           // (N,3)

    const int N    = in_sizes[0] / 2;
    const int grid = (N + PTS_PER_BLOCK - 1) / PTS_PER_BLOCK;
    SphericalHarmonics_kernel<<<grid, BLOCK_THREADS, 0, stream>>>(dirs, dc, rest, out, N);
}